// downprompt_75496935129292
// MI455X (gfx1250) — compile-verified
//
#include <hip/hip_runtime.h>
#include <math.h>

#define H   256
#define K1  16
#define K2  32
#define KT  48
#define NC  7
#define CEPS 1e-8f
#define FMAXV 3.4028234663852886e38f

typedef __attribute__((ext_vector_type(2))) float v2f;
typedef __attribute__((ext_vector_type(8))) float v8f;

__device__ __forceinline__ float wred_sum(float v) {
#pragma unroll
    for (int m = 16; m >= 1; m >>= 1) v += __shfl_xor(v, m, 32);
    return v;
}

__device__ __forceinline__ float nan2num(float v) {
    if (v != v) return 0.0f;                       // NaN -> 0
    if (v ==  __builtin_inff()) return  FMAXV;     // +inf -> FLT_MAX
    if (v == -__builtin_inff()) return -FMAXV;     // -inf -> -FLT_MAX
    return v;
}

// Load/store 8 consecutive floats as two b128 ops.
__device__ __forceinline__ void ld8(const float* __restrict__ p, float v[8]) {
    float4 a = *reinterpret_cast<const float4*>(p);
    float4 b = *reinterpret_cast<const float4*>(p + 4);
    v[0]=a.x; v[1]=a.y; v[2]=a.z; v[3]=a.w;
    v[4]=b.x; v[5]=b.y; v[6]=b.z; v[7]=b.w;
}
__device__ __forceinline__ void st8(float* __restrict__ p, const float v[8]) {
    *reinterpret_cast<float4*>(p)     = make_float4(v[0], v[1], v[2], v[3]);
    *reinterpret_cast<float4*>(p + 4) = make_float4(v[4], v[5], v[6], v[7]);
}

__global__ void zero_kernel(float* p, int n) {
    int i = blockIdx.x * blockDim.x + threadIdx.x;
    if (i < n) p[i] = 0.0f;
}

__device__ __forceinline__ float sim_one(const float* __restrict__ rowp,
                                         const float w[8], const float c[8],
                                         float cnorm) {
    float v[8];
    ld8(rowp, v);
    float dot = 0.0f, nn2 = 0.0f;
#pragma unroll
    for (int j = 0; j < 8; j++) {
        float x = nan2num(v[j] * w[j]);
        dot += x * c[j];
        nn2 += x * x;
    }
    dot = wred_sum(dot);
    nn2 = wred_sum(nn2);
    return dot / (cnorm * fmaxf(sqrtf(nn2), CEPS));
}

__device__ __forceinline__ void acc_one(const float* __restrict__ rowp,
                                        const float w[8], float e, float acc[8]) {
    float v[8];
    ld8(rowp, v);
#pragma unroll
    for (int j = 0; j < 8; j++) acc[j] += e * nan2num(v[j] * w[j]);
}

// One wave32 per row b: center, 48 cosine sims, softmax-weighted neighbor sum.
// Lane-contiguous layout h = lane*8 + j -> two global_load_b128 per row gather.
__global__ void __launch_bounds__(256) rawret_kernel(
    const float* __restrict__ embeds, const float* __restrict__ w_self,
    const float* __restrict__ w_nb, const float* __restrict__ w_nb2,
    const int* __restrict__ idx, const int* __restrict__ nbr,
    const int* __restrict__ nbr2,
    float* __restrict__ rawret, float* __restrict__ rn, int B)
{
    int t = threadIdx.x;
    int wave = t >> 5, lane = t & 31;
    int b = blockIdx.x * 8 + wave;
    if (b >= B) return;
    int h0 = lane * 8;

    // Weight vectors: each lane only ever needs its own 8 positions -> registers.
    float ws[8], wn[8], wn2[8];
    ld8(w_self + h0, ws);
    ld8(w_nb   + h0, wn);
    ld8(w_nb2  + h0, wn2);

    long long crow = (long long)idx[b] * H;
    float c[8], ev[8];
    ld8(embeds + crow + h0, ev);
    float cn2 = 0.0f;
#pragma unroll
    for (int j = 0; j < 8; j++) { c[j] = ev[j] * ws[j]; cn2 += c[j] * c[j]; }
    cn2 = wred_sum(cn2);
    float cnorm = fmaxf(sqrtf(cn2), CEPS);

    // Pass 1: cosine sims; sim k kept on lane k (s0) / lane k-32 (s1).
    float s0 = 0.0f, s1 = 0.0f, smax = -3.0e38f;
    for (int k = 0; k < K1; k++) {
        long long nrow = (long long)nbr[b * K1 + k] * H;
        float sim = sim_one(embeds + nrow + h0, wn, c, cnorm);
        smax = fmaxf(smax, sim);
        if (lane == k) s0 = sim;
    }
    for (int k = 0; k < K2; k++) {
        int kk = k + K1;
        long long nrow = (long long)nbr2[b * K2 + k] * H;
        float sim = sim_one(embeds + nrow + h0, wn2, c, cnorm);
        smax = fmaxf(smax, sim);
        if (kk < 32) { if (lane == kk)      s0 = sim; }
        else         { if (lane == kk - 32) s1 = sim; }
    }

    // Pass 2: softmax weights + weighted sum (re-gather, L2-resident).
    float acc[8];
#pragma unroll
    for (int j = 0; j < 8; j++) acc[j] = 0.0f;
    float denom = 0.0f;
    for (int k = 0; k < K1; k++) {
        float e = expf(__shfl(s0, k, 32) - smax);
        denom += e;
        long long nrow = (long long)nbr[b * K1 + k] * H;
        acc_one(embeds + nrow + h0, wn, e, acc);
    }
    for (int k = 0; k < K2; k++) {
        int kk = k + K1;
        float sim = (kk < 32) ? __shfl(s0, kk, 32) : __shfl(s1, kk - 32, 32);
        float e = expf(sim - smax);
        denom += e;
        long long nrow = (long long)nbr2[b * K2 + k] * H;
        acc_one(embeds + nrow + h0, wn2, e, acc);
    }

    float o[8];
    float rr2 = 0.0f;
#pragma unroll
    for (int j = 0; j < 8; j++) {
        o[j] = acc[j] / denom + c[j];
        rr2 += o[j] * o[j];
    }
    st8(rawret + (long long)b * H + h0, o);
    rr2 = wred_sum(rr2);
    if (lane == 0) rn[b] = fmaxf(sqrtf(rr2), CEPS);
}

// Per-class column sums + label histogram.
__global__ void __launch_bounds__(256) classsum_kernel(
    const float* __restrict__ rawret, const int* __restrict__ labels,
    float* __restrict__ classSums, float* __restrict__ counts,
    int B, int rowsPerBlk)
{
    int h = threadIdx.x;
    int r0 = blockIdx.x * rowsPerBlk;
    int r1 = min(r0 + rowsPerBlk, B);
    float acc[NC];
#pragma unroll
    for (int cc = 0; cc < NC; cc++) acc[cc] = 0.0f;
    for (int r = r0; r < r1; r++) {
        int lbl = labels[r];
        float v = rawret[(long long)r * H + h];
#pragma unroll
        for (int cc = 0; cc < NC; cc++) acc[cc] += (lbl == cc) ? v : 0.0f;
    }
#pragma unroll
    for (int cc = 0; cc < NC; cc++) atomicAdd(&classSums[cc * H + h], acc[cc]);
    if (h < NC) {
        float cnt = 0.0f;
        for (int r = r0; r < r1; r++) cnt += (labels[r] == h) ? 1.0f : 0.0f;
        atomicAdd(&counts[h], cnt);
    }
}

// ave = sums/max(counts,1); build transposed padded B matrix [256][16] + norms.
__global__ void __launch_bounds__(256) ave_kernel(
    const float* __restrict__ classSums, const float* __restrict__ counts,
    float* __restrict__ avePadT, float* __restrict__ an)
{
    __shared__ float ns[NC];
    int t = threadIdx.x;
    if (t < NC) ns[t] = 0.0f;
    __syncthreads();
#pragma unroll
    for (int cc = 0; cc < NC; cc++) {
        float v = classSums[cc * H + t] / fmaxf(counts[cc], 1.0f);
        avePadT[t * 16 + cc] = v;
        atomicAdd(&ns[cc], v * v);
    }
#pragma unroll
    for (int cc = NC; cc < 16; cc++) avePadT[t * 16 + cc] = 0.0f;
    __syncthreads();
    if (t < NC)      an[t] = fmaxf(sqrtf(ns[t]), CEPS);
    else if (t < 16) an[t] = 1.0f;
}

// WMMA GEMM: [16,256] x [256,16(pad of 7)] per wave, then row softmax.
__global__ void __launch_bounds__(256) gemm_softmax_kernel(
    const float* __restrict__ rawret, const float* __restrict__ avePadT,
    const float* __restrict__ rn, const float* __restrict__ an,
    float* __restrict__ out, int ntiles, int B)
{
    int wave = threadIdx.x >> 5, lane = threadIdx.x & 31;
    int tile = blockIdx.x * 8 + wave;
    if (tile >= ntiles) return;

    int half = lane >> 4, m16 = lane & 15;
    // A fragment: lanes 0-15 hold K = k0,k0+1 ; lanes 16-31 hold K = k0+2,k0+3
    int arow = tile * 16 + m16;
    int arowc = min(arow, B - 1);
    const float* ap = rawret + (long long)arowc * H;
    int koff = half * 2;

    v8f acc = {};
    for (int k0 = 0; k0 < H; k0 += 4) {
        v2f a, bf;
        a.x  = ap[k0 + koff];
        a.y  = ap[k0 + koff + 1];
        bf.x = avePadT[(k0 + koff) * 16 + m16];
        bf.y = avePadT[(k0 + koff + 1) * 16 + m16];
        acc = __builtin_amdgcn_wmma_f32_16x16x4_f32(
            /*neg_a=*/false, a, /*neg_b=*/false, bf,
            /*c_mod=*/(short)0, acc, /*reuse_a=*/false, /*reuse_b=*/false);
    }

    // D layout: VGPR j -> row j (lanes 0-15) / row j+8 (lanes 16-31), col = lane&15
#pragma unroll
    for (int j = 0; j < 8; j++) {
        int m = j + 8 * half;
        int grow = tile * 16 + m;
        int growc = min(grow, B - 1);
        float scale = rn[growc] * an[m16];
        float val = acc[j] / scale;
        float s = (m16 < NC) ? val : -3.0e38f;
        float mx = s;
#pragma unroll
        for (int d = 1; d < 16; d <<= 1) mx = fmaxf(mx, __shfl_xor(mx, d, 32));
        float e = expf(s - mx);
        float sm = e;
#pragma unroll
        for (int d = 1; d < 16; d <<= 1) sm += __shfl_xor(sm, d, 32);
        if (m16 < NC && grow < B) out[(long long)grow * NC + m16] = e / sm;
    }
}

extern "C" void kernel_launch(void* const* d_in, const int* in_sizes, int n_in,
                              void* d_out, int out_size, void* d_ws, size_t ws_size,
                              hipStream_t stream)
{
    const float* embeds = (const float*)d_in[0];
    const float* w_self = (const float*)d_in[1];
    const float* w_nb   = (const float*)d_in[2];
    const float* w_nb2  = (const float*)d_in[3];
    const int*   idx    = (const int*)d_in[4];
    const int*   labels = (const int*)d_in[5];
    const int*   nbrs   = (const int*)d_in[6];
    const int*   nbrs2  = (const int*)d_in[7];
    int B = in_sizes[4];
    float* out = (float*)d_out;

    float* ws        = (float*)d_ws;
    float* rawret    = ws;                         // B*H
    float* rn        = rawret + (size_t)B * H;     // B
    float* classSums = rn + B;                     // NC*H
    float* counts    = classSums + NC * H;         // NC (padded to 16)
    float* avePadT   = counts + 16;                // H*16
    float* an        = avePadT + H * 16;           // 16

    // Zero class accumulators (contiguous: classSums + counts pad) every call.
    zero_kernel<<<(NC * H + 16 + 255) / 256, 256, 0, stream>>>(classSums, NC * H + 16);

    int nwBlocks = (B + 7) / 8;
    rawret_kernel<<<nwBlocks, 256, 0, stream>>>(embeds, w_self, w_nb, w_nb2,
                                                idx, nbrs, nbrs2, rawret, rn, B);

    const int NBLK = 56;
    int rpb = (B + NBLK - 1) / NBLK;
    classsum_kernel<<<NBLK, 256, 0, stream>>>(rawret, labels, classSums, counts, B, rpb);

    ave_kernel<<<1, 256, 0, stream>>>(classSums, counts, avePadT, an);

    int ntiles = (B + 15) / 16;
    gemm_softmax_kernel<<<(ntiles + 7) / 8, 256, 0, stream>>>(rawret, avePadT, rn, an,
                                                              out, ntiles, B);
}